// MLP_HENN_35862976921650
// MI455X (gfx1250) — compile-verified
//
#include <hip/hip_runtime.h>
#include <hip/hip_bf16.h>
#include <math.h>

typedef float v2f __attribute__((ext_vector_type(2)));
typedef float v8f __attribute__((ext_vector_type(8)));

#define DIMD 256          // feature dim D
#define DIMH 256          // hidden dim H
#define ZPITCH 260        // padded LDS row pitch (4-dword skew -> conflict-free A reads)
#define TILE_E 16         // edges per block

__launch_bounds__(256, 2)
__global__ void henn_fused_kernel(const float* __restrict__ x,
                                  const int*   __restrict__ tnodes,
                                  const int*   __restrict__ tids,
                                  const float* __restrict__ W1,
                                  const float* __restrict__ b1,
                                  const float* __restrict__ W2,
                                  const float* __restrict__ b2,
                                  float*       __restrict__ out,
                                  int P, int E)
{
    __shared__ float zs[TILE_E * ZPITCH];   // 16 x 256 Z tile (padded)  ~16.6 KB
    __shared__ float part[8][32][9];        // per-wave/lane partial dots ~9.2 KB

    const int tid   = threadIdx.x;
    const int wave  = tid >> 5;
    const int lane  = tid & 31;
    const int ebase = blockIdx.x * TILE_E;

    // ---------------- Phase A: gather + segment-sum (2 edges per wave) ----------------
    for (int rr = 0; rr < 2; ++rr) {
        const int row = wave + rr * 8;        // 0..15
        const int e   = ebase + row;

        // lower_bound(tids, e) and lower_bound(tids, e+1): wave-uniform (redundant per lane)
        int lo = 0, hi = P;
        while (lo < hi) { int m = (lo + hi) >> 1; if (tids[m] <  e) lo = m + 1; else hi = m; }
        int lo2 = lo, hi2 = P;
        while (lo2 < hi2) { int m = (lo2 + hi2) >> 1; if (tids[m] <= e) lo2 = m + 1; else hi2 = m; }

        float4 a0 = make_float4(0.f, 0.f, 0.f, 0.f);
        float4 a1 = make_float4(0.f, 0.f, 0.f, 0.f);
        for (int p = lo; p < lo2; ++p) {
            const int node = tnodes[p];                      // uniform broadcast load
            const float4* xr = (const float4*)(x + (size_t)node * DIMD);
            float4 v0 = xr[lane];          // dims lane*4 .. lane*4+3      (coalesced 512B)
            float4 v1 = xr[lane + 32];     // dims 128+lane*4 ..           (coalesced 512B)
            a0.x += v0.x; a0.y += v0.y; a0.z += v0.z; a0.w += v0.w;
            a1.x += v1.x; a1.y += v1.y; a1.z += v1.z; a1.w += v1.w;
        }
        float* zrow = zs + row * ZPITCH;
        *(float4*)(zrow + lane * 4)       = a0;
        *(float4*)(zrow + 128 + lane * 4) = a1;
    }
    __syncthreads();

    // ---------------- Phase B: [16 x 256] @ W1^T via V_WMMA_F32_16X16X4_F32 ----------------
    // Each wave owns two 16-wide H tiles: h0 = wave*32, h1 = wave*32+16.
    const int h0    = wave * 32;
    const int h1    = h0 + 16;
    const int col   = lane & 15;                 // N index within tile; also A-matrix row M
    const int khalf = (lane < 16) ? 0 : 2;       // fp32 16x16x4 A/B layout K-split

    const float* arow  = zs + col * ZPITCH + khalf;              // A: Z[M=col][k+khalf, +1]
    const float* brow0 = W1 + (size_t)(h0 + col) * DIMD + khalf; // B: W1[h][k+khalf, +1]
    const float* brow1 = W1 + (size_t)(h1 + col) * DIMD + khalf;

    v8f c0 = {0.f, 0.f, 0.f, 0.f, 0.f, 0.f, 0.f, 0.f};
    v8f c1 = {0.f, 0.f, 0.f, 0.f, 0.f, 0.f, 0.f, 0.f};
    #pragma unroll 8
    for (int k = 0; k < DIMD; k += 4) {
        v2f a  = *(const v2f*)(arow + k);     // ds_load_b64, bank-conflict-free
        v2f bb0 = *(const v2f*)(brow0 + k);   // global_load_b64, WGP$-resident W1
        v2f bb1 = *(const v2f*)(brow1 + k);
        c0 = __builtin_amdgcn_wmma_f32_16x16x4_f32(false, a, false, bb0, (short)0, c0, false, false);
        c1 = __builtin_amdgcn_wmma_f32_16x16x4_f32(false, a, false, bb1, (short)0, c1, false, false);
    }

    // ---------------- Epilogue: bias + relu + dot(W2) fused in registers ----------------
    // C/D layout: VGPR v, lanes 0-15 -> (M=v,   N=lane); lanes 16-31 -> (M=v+8, N=lane-16)
    const float w2a = W2[h0 + col];
    const float w2b = W2[h1 + col];
    const float b1a = b1[h0 + col];
    const float b1b = b1[h1 + col];
    #pragma unroll
    for (int v = 0; v < 8; ++v) {
        float r0 = fmaxf(c0[v] + b1a, 0.0f);
        float r1 = fmaxf(c1[v] + b1b, 0.0f);
        part[wave][lane][v] = fmaf(r0, w2a, r1 * w2b);
    }
    __syncthreads();

    // ---------------- Final cross-wave reduction, sigmoid, store ----------------
    if (tid < TILE_E) {
        const int r     = tid;
        const int lbase = (r < 8) ? 0 : 16;   // rows 0-7 live in lanes 0-15, rows 8-15 in 16-31
        const int slot  = r & 7;
        float s = b2[0];
        #pragma unroll
        for (int w = 0; w < 8; ++w)
            #pragma unroll
            for (int l = 0; l < 16; ++l)
                s += part[w][lbase + l][slot];
        const int e = ebase + r;
        if (e < E) out[e] = 1.0f / (1.0f + expf(-s));
    }
}

extern "C" void kernel_launch(void* const* d_in, const int* in_sizes, int n_in,
                              void* d_out, int out_size, void* d_ws, size_t ws_size,
                              hipStream_t stream) {
    const float* x   = (const float*)d_in[0];
    const int*   tn  = (const int*)  d_in[1];
    const int*   ti  = (const int*)  d_in[2];
    const float* W1  = (const float*)d_in[3];
    const float* b1v = (const float*)d_in[4];
    const float* W2  = (const float*)d_in[5];
    const float* b2v = (const float*)d_in[6];
    float* out = (float*)d_out;

    const int P = in_sizes[1];     // number of incidence pairs
    const int E = out_size;        // number of hyperedges
    const int blocks = (E + TILE_E - 1) / TILE_E;

    henn_fused_kernel<<<blocks, 256, 0, stream>>>(x, tn, ti, W1, b1v, W2, b2v, out, P, E);
}